// Blur_46110768890056
// MI455X (gfx1250) — compile-verified
//
#include <hip/hip_runtime.h>
#include <hip/hip_bf16.h>

// Depthwise 4x4 FIR blur (upfirdn2d, pad=(2,2)), fp32, NCHW.
// x: (8, 256, 256, 256) -> out: (8, 256, 257, 257)
//
// Memory-bound (~1.08 GB traffic -> ~46 us at 23.3 TB/s on MI455X). Strategy:
//  - 64x64 output tile per 256-thread WG (8 waves)
//  - input tile staged in LDS via gfx1250 async copies:
//      global_load_async_to_lds_b128 (16B/lane/issue, ASYNCcnt tracked)
//    tile realigned to global 16B boundaries (cols gx0-4 .. gx0+67, stride 72)
//    so every 4-col group is fully in- or out-of-bounds; zero-filled LDS
//    provides the exact zero padding for skipped groups.
//  - separable 4-tap H pass then V pass (16 -> ~8.2 MAC/output keeps fp32
//    VALU time well under the HBM time)
//  - coalesced non-temporal fp32 stores (output is write-once streaming)

#define TILE     64
#define INROWS   67   // input tile rows (TILE + kh - 1)
#define INSTRIDE 72   // LDS input row stride in floats (16B-aligned, 18 groups)
#define NGRP     18   // 4-col groups per input row
#define HSTRIDE  64

__global__ __launch_bounds__(256) void Blur_46110768890056_kernel(
    const float* __restrict__ x, const float* __restrict__ kern,
    float* __restrict__ out) {
  constexpr int C = 256, H = 256, W = 256, OH = 257, OW = 257;
  constexpr int NTX = 5;  // ceil(257/64)

  __shared__ float sIn[INROWS * INSTRIDE];  // 67*72 floats = 18.8 KB
  __shared__ float sH[INROWS * HSTRIDE];    // 67*64 floats = 16.8 KB

  const int tid = threadIdx.x;
  const int tx = blockIdx.x % NTX;
  const int ty = blockIdx.x / NTX;
  const int gx0 = tx * TILE;
  const int gy0 = ty * TILE;
  const int chan = blockIdx.y;
  const int batch = blockIdx.z;

  const float* __restrict__ img = x + ((size_t)(batch * C + chan) << 16);  // H*W = 65536

  // Separable weights from the rank-1 2D FIR buffer.
  // True convolution: w[i][j] = K[3-i][3-j] = R[i] * Cw[j] with
  //   Cw[j] = K[0][3-j],  R[i] = K[3-i][0] / K[0][0]
  const float k0 = kern[0], k1 = kern[1], k2 = kern[2], k3 = kern[3];
  const float k4 = kern[4], k8 = kern[8], k12 = kern[12];
  const float cw0 = k3, cw1 = k2, cw2 = k1, cw3 = k0;
  const float inv00 = 1.0f / k0;
  const float r0 = k12 * inv00, r1 = k8 * inv00, r2 = k4 * inv00, r3 = 1.0f;

  // ---- Phase 0: zero the LDS input tile (implements zero padding) ----
  {
    const float4 z = make_float4(0.f, 0.f, 0.f, 0.f);
    for (int i = tid; i < (INROWS * INSTRIDE) / 4; i += 256)
      *reinterpret_cast<float4*>(&sIn[i * 4]) = z;
  }
  __syncthreads();

  // ---- Phase 1: async global->LDS b128 copy of the valid input region ----
  // LDS col c' holds global col (gx0 - 4 + c'); both sides 16B aligned.
  for (int l = tid; l < INROWS * NGRP; l += 256) {
    const int r = l / NGRP;
    const int g = l - r * NGRP;
    const int gr = gy0 - 2 + r;
    const int gc0 = gx0 - 4 + g * 4;  // group fully in or fully out (W%4==0)
    if ((unsigned)gr < (unsigned)H && (unsigned)gc0 <= (unsigned)(W - 4)) {
      const float* gp = img + (gr << 8) + gc0;
      // low 32 bits of the flat LDS pointer == LDS byte offset
      const unsigned lofs =
          (unsigned)(unsigned long long)(uintptr_t)(&sIn[r * INSTRIDE + g * 4]);
      asm volatile("global_load_async_to_lds_b128 %0, %1, off"
                   :
                   : "v"(lofs), "v"((unsigned long long)(uintptr_t)gp)
                   : "memory");
    }
  }
  asm volatile("s_wait_asynccnt 0" ::: "memory");
  __syncthreads();

  // ---- Phase 2: horizontal 4-tap pass ----
  // Output col c (0..63) needs LDS input cols c+2 .. c+5.
  for (int t = tid; t < INROWS * (TILE / 4); t += 256) {
    const int rrow = t >> 4;  // 0..66
    const int g = t & 15;     // 4-wide output column group
    const int base = rrow * INSTRIDE + g * 4;
    const float4 p0 = *reinterpret_cast<const float4*>(&sIn[base]);      // cols 4g..4g+3
    const float4 p1 = *reinterpret_cast<const float4*>(&sIn[base + 4]);  // cols 4g+4..4g+7
    const float e8 = sIn[base + 8];                                      // col  4g+8
    float4 hv;
    hv.x = cw0 * p0.z + cw1 * p0.w + cw2 * p1.x + cw3 * p1.y;
    hv.y = cw0 * p0.w + cw1 * p1.x + cw2 * p1.y + cw3 * p1.z;
    hv.z = cw0 * p1.x + cw1 * p1.y + cw2 * p1.z + cw3 * p1.w;
    hv.w = cw0 * p1.y + cw1 * p1.z + cw2 * p1.w + cw3 * e8;
    *reinterpret_cast<float4*>(&sH[rrow * HSTRIDE + g * 4]) = hv;
  }
  __syncthreads();

  // ---- Phase 3: vertical 4-tap pass + coalesced non-temporal stores ----
  float* __restrict__ oimg = out + (size_t)(batch * C + chan) * (size_t)(OH * OW);
  for (int t = tid; t < TILE * (TILE / 4); t += 256) {
    const int c = t & 63;         // column within tile; lanes -> consecutive
    const int y0 = (t >> 6) * 4;  // 4-row run: 0,4,...,60
    const int ox = gx0 + c;
    const float s0 = sH[(y0 + 0) * HSTRIDE + c];
    const float s1 = sH[(y0 + 1) * HSTRIDE + c];
    const float s2 = sH[(y0 + 2) * HSTRIDE + c];
    const float s3 = sH[(y0 + 3) * HSTRIDE + c];
    const float s4 = sH[(y0 + 4) * HSTRIDE + c];
    const float s5 = sH[(y0 + 5) * HSTRIDE + c];
    const float s6 = sH[(y0 + 6) * HSTRIDE + c];
    const float o0 = r0 * s0 + r1 * s1 + r2 * s2 + r3 * s3;
    const float o1 = r0 * s1 + r1 * s2 + r2 * s3 + r3 * s4;
    const float o2 = r0 * s2 + r1 * s3 + r2 * s4 + r3 * s5;
    const float o3 = r0 * s3 + r1 * s4 + r2 * s5 + r3 * s6;
    if (ox < OW) {
      const int oy = gy0 + y0;
      if (oy + 3 < OH) {
        __builtin_nontemporal_store(o0, &oimg[(oy + 0) * OW + ox]);
        __builtin_nontemporal_store(o1, &oimg[(oy + 1) * OW + ox]);
        __builtin_nontemporal_store(o2, &oimg[(oy + 2) * OW + ox]);
        __builtin_nontemporal_store(o3, &oimg[(oy + 3) * OW + ox]);
      } else {
        if (oy + 0 < OH) __builtin_nontemporal_store(o0, &oimg[(oy + 0) * OW + ox]);
        if (oy + 1 < OH) __builtin_nontemporal_store(o1, &oimg[(oy + 1) * OW + ox]);
        if (oy + 2 < OH) __builtin_nontemporal_store(o2, &oimg[(oy + 2) * OW + ox]);
        if (oy + 3 < OH) __builtin_nontemporal_store(o3, &oimg[(oy + 3) * OW + ox]);
      }
    }
  }
}

extern "C" void kernel_launch(void* const* d_in, const int* in_sizes, int n_in,
                              void* d_out, int out_size, void* d_ws, size_t ws_size,
                              hipStream_t stream) {
  (void)in_sizes; (void)n_in; (void)out_size; (void)d_ws; (void)ws_size;
  const float* x = (const float*)d_in[0];     // (8,256,256,256) fp32
  const float* kern = (const float*)d_in[1];  // (4,4) fp32 normalized FIR
  float* out = (float*)d_out;                 // (8,256,257,257) fp32

  dim3 grid(5 * 5 /* 64x64 tiles over 257x257 */, 256 /* C */, 8 /* B */);
  Blur_46110768890056_kernel<<<grid, dim3(256), 0, stream>>>(x, kern, out);
}